// UserModel_28999619182733
// MI455X (gfx1250) — compile-verified
//
#include <hip/hip_runtime.h>
#include <hip/hip_bf16.h>

typedef __attribute__((ext_vector_type(2))) float v2f;
typedef __attribute__((ext_vector_type(8))) float v8f;

#define BQ 16
#define SQ 512
#define VQ 128
#define NCQ 10000

__device__ __forceinline__ v8f wmma4(v2f a, v2f b, v8f c) {
  // D = A(16x4,f32) * B(4x16,f32) + C(16x16,f32)
  return __builtin_amdgcn_wmma_f32_16x16x4_f32(false, a, false, b, (short)0, c,
                                               false, false);
}

__device__ __forceinline__ float sigmoidf_(float x) {
  return 1.f / (1.f + __expf(-x));
}

// ---------------- Kernel A: build X = [D2_w[d], r*v_r], gamma = D1_w[d] ----
__global__ __launch_bounds__(256) void k_embed(const int* __restrict__ d_seq,
                                               const float* __restrict__ r_seq,
                                               const float* __restrict__ D1,
                                               const float* __restrict__ D2,
                                               const float* __restrict__ v_r,
                                               float* __restrict__ X,
                                               float* __restrict__ gamma_out) {
  int wave = threadIdx.x >> 5;
  int lane = threadIdx.x & 31;
  int row = blockIdx.x * 8 + wave;               // 1024 blocks * 8 rows
  int d = d_seq[row];
  float r = r_seq[row];
  float4 vd = ((const float4*)(D2 + (size_t)d * VQ))[lane];
  ((float4*)(X + (size_t)row * 256))[lane] = vd;
  float4 v = ((const float4*)v_r)[lane];
  float4 o;
  o.x = r * v.x; o.y = r * v.y; o.z = r * v.z; o.w = r * v.w;
  ((float4*)(X + (size_t)row * 256 + 128))[lane] = o;
  if (lane == 0) gamma_out[row] = D1[d];
}

// ---------------- zero helper ---------------------------------------------
__global__ void k_zero(float* __restrict__ p, int n) {
  int i = blockIdx.x * 256 + threadIdx.x;
  if (i < n) p[i] = 0.f;
}

// ---------------- Kernel B: XP = X @ W_ih^T + b_ih  (8192x256x384) --------
__global__ __launch_bounds__(256) void k_gemm_xp(const float* __restrict__ X,
                                                 const float* __restrict__ W_ih,
                                                 const float* __restrict__ b_ih,
                                                 float* __restrict__ XP) {
  int wave = threadIdx.x >> 5;
  int lane = threadIdx.x & 31;
  int hl = lane >> 4, lr = lane & 15;
  int tile = blockIdx.x * 8 + wave;              // 12288 tiles
  int rt = tile / 24, ct = tile % 24;
  int row0 = rt * 16, col0 = ct * 16;
  const v2f* Arow = (const v2f*)(X + (size_t)(row0 + lr) * 256);
  const v2f* Brow = (const v2f*)(W_ih + (size_t)(col0 + lr) * 256);
  v8f acc = {0.f, 0.f, 0.f, 0.f, 0.f, 0.f, 0.f, 0.f};
#pragma unroll 8
  for (int k0 = 0; k0 < 256; k0 += 4)
    acc = wmma4(Arow[(k0 >> 1) + hl], Brow[(k0 >> 1) + hl], acc);
  float bias = b_ih[col0 + lr];
#pragma unroll
  for (int i = 0; i < 8; i++)
    XP[(size_t)(row0 + i + 8 * hl) * 384 + col0 + lr] = acc[i] + bias;
}

// ---------------- Kernel C: GRU recurrence (single WG, 16 waves) ----------
#define HS 132   // padded stride for [16,128] LDS tiles (bank-conflict free)
#define PS 388   // padded stride for [16,384] LDS tiles
__global__ __launch_bounds__(512) void k_gru(const float* __restrict__ XP,
                                             const float* __restrict__ W_hh,
                                             const float* __restrict__ b_hh,
                                             float* __restrict__ h_out) {
  __shared__ float h[16 * HS];
  __shared__ float hp[16 * PS];
  int tid = threadIdx.x;
  int wave = tid >> 5, lane = tid & 31, hl = lane >> 4, lr = lane & 15;
  for (int i = tid; i < 16 * HS; i += 512) h[i] = 0.f;
  __syncthreads();
  for (int t = 0; t < SQ; ++t) {
    // hp = h @ W_hh^T + b_hh  (16x384, 24 col-tiles over 16 waves)
    for (int ct = wave; ct < 24; ct += 16) {
      int col0 = ct * 16;
      const v2f* Arow = (const v2f*)(h + lr * HS);
      const v2f* Brow = (const v2f*)(W_hh + (size_t)(col0 + lr) * VQ);
      v8f acc = {0.f, 0.f, 0.f, 0.f, 0.f, 0.f, 0.f, 0.f};
#pragma unroll 8
      for (int k0 = 0; k0 < VQ; k0 += 4)
        acc = wmma4(Arow[(k0 >> 1) + hl], Brow[(k0 >> 1) + hl], acc);
      float bias = b_hh[col0 + lr];
#pragma unroll
      for (int i = 0; i < 8; i++)
        hp[(i + 8 * hl) * PS + col0 + lr] = acc[i] + bias;
    }
    __syncthreads();
    // gates: 2048 state elements over 512 threads
#pragma unroll
    for (int rep = 0; rep < 4; ++rep) {
      int e = tid + rep * 512;
      int b = e >> 7, v = e & 127;
      size_t row = (size_t)(b * SQ + t) * 384;
      float xr = XP[row + v], xz = XP[row + VQ + v], xn = XP[row + 2 * VQ + v];
      float hr = hp[b * PS + v], hz = hp[b * PS + VQ + v], hn = hp[b * PS + 2 * VQ + v];
      float rg = sigmoidf_(xr + hr);
      float zg = sigmoidf_(xz + hz);
      float ng = tanhf(xn + rg * hn);
      float hprev = h[b * HS + v];
      float hnew = (1.f - zg) * ng + zg * hprev;
      h[b * HS + v] = hnew;
      h_out[(size_t)(b * SQ + t) * VQ + v] = hnew;
    }
    __syncthreads();
  }
}

// ---------------- Kernel D: alpha = MLP1(h_seq)  ---------------------------
__global__ __launch_bounds__(128) void k_alpha(
    const float* __restrict__ h_seq, const float* __restrict__ W1,
    const float* __restrict__ b1, const float* __restrict__ W2,
    const float* __restrict__ b2, const float* __restrict__ W3,
    const float* __restrict__ b3, float* __restrict__ alpha_out) {
  __shared__ float buf[4][16 * HS];
  int wave = threadIdx.x >> 5, lane = threadIdx.x & 31;
  int hl = lane >> 4, lr = lane & 15;
  int row0 = (blockIdx.x * 4 + wave) * 16;       // 128 blocks * 4 tiles
  float* wbuf = &buf[wave][0];
  // layer 1: h(global) @ W1[in,out] -> relu -> wbuf
#pragma unroll
  for (int nt = 0; nt < 8; ++nt) {
    int col0 = nt * 16;
    const v2f* Arow = (const v2f*)(h_seq + (size_t)(row0 + lr) * VQ);
    v8f acc = {0.f, 0.f, 0.f, 0.f, 0.f, 0.f, 0.f, 0.f};
#pragma unroll 8
    for (int k0 = 0; k0 < VQ; k0 += 4) {
      int k = k0 + 2 * hl;
      v2f a = Arow[(k0 >> 1) + hl];
      v2f b;
      b.x = W1[(size_t)k * VQ + col0 + lr];
      b.y = W1[(size_t)(k + 1) * VQ + col0 + lr];
      acc = wmma4(a, b, acc);
    }
    float bias = b1[col0 + lr];
#pragma unroll
    for (int i = 0; i < 8; i++) {
      float vv = acc[i] + bias;
      wbuf[(i + 8 * hl) * HS + col0 + lr] = vv > 0.f ? vv : 0.f;
    }
  }
  __syncthreads();
  // layer 2: keep results in registers, then overwrite wbuf
  v8f res[8];
#pragma unroll
  for (int nt = 0; nt < 8; ++nt) {
    int col0 = nt * 16;
    const v2f* Arow = (const v2f*)(wbuf + lr * HS);
    v8f acc = {0.f, 0.f, 0.f, 0.f, 0.f, 0.f, 0.f, 0.f};
#pragma unroll 8
    for (int k0 = 0; k0 < VQ; k0 += 4) {
      int k = k0 + 2 * hl;
      v2f a = Arow[(k0 >> 1) + hl];
      v2f b;
      b.x = W2[(size_t)k * VQ + col0 + lr];
      b.y = W2[(size_t)(k + 1) * VQ + col0 + lr];
      acc = wmma4(a, b, acc);
    }
    float bias = b2[col0 + lr];
#pragma unroll
    for (int i = 0; i < 8; i++) {
      float vv = acc[i] + bias;
      res[nt][i] = vv > 0.f ? vv : 0.f;
    }
  }
  __syncthreads();
#pragma unroll
  for (int nt = 0; nt < 8; ++nt) {
    int col0 = nt * 16;
#pragma unroll
    for (int i = 0; i < 8; i++) wbuf[(i + 8 * hl) * HS + col0 + lr] = res[nt][i];
  }
  __syncthreads();
  // layer 3: dot with W3[128,1]
  float p = 0.f;
  const float* hrow = wbuf + lr * HS + hl * 64;
#pragma unroll 8
  for (int k = 0; k < 64; ++k) p += hrow[k] * W3[hl * 64 + k];
  p += __shfl_xor(p, 16, 32);
  if (hl == 0) alpha_out[row0 + lr] = p + b3[0];
}

// ---------------- Kernel E: beta scan (single WG, 8 waves) ----------------
__global__ __launch_bounds__(256) void k_beta(
    const int* __restrict__ c_seq, const float* __restrict__ X,
    const float* __restrict__ v_beta, const float* __restrict__ W1,
    const float* __restrict__ b1, const float* __restrict__ W2,
    const float* __restrict__ b2, const float* __restrict__ W3,
    const float* __restrict__ b3, float* __restrict__ Cstate,
    float* __restrict__ beta_out) {
  __shared__ float y[16 * PS];
  __shared__ float g1[16 * HS];
  __shared__ float g2[16 * HS];
  __shared__ int cidx[16];
  int tid = threadIdx.x;
  int wave = tid >> 5, lane = tid & 31, hl = lane >> 4, lr = lane & 15;
  for (int t = 0; t < SQ; ++t) {
    if (tid < 16) cidx[tid] = c_seq[tid * SQ + t];
    __syncthreads();
    // y = [beta_prev * v_beta, v_d, v_r]   (16 x 384)
    for (int e = tid; e < 16 * 384; e += 256) {
      int b = e / 384, j = e - b * 384;
      float val;
      if (j < VQ)
        val = Cstate[(size_t)b * NCQ + cidx[b]] * v_beta[j];
      else
        val = X[(size_t)(b * SQ + t) * 256 + (j - VQ)];
      y[b * PS + j] = val;
    }
    __syncthreads();
    // layer 1: y(16x384) @ W1[384,128] -> relu -> g1 ; wave -> col-tile
    {
      int col0 = wave * 16;
      const v2f* Arow = (const v2f*)(y + lr * PS);
      v8f acc = {0.f, 0.f, 0.f, 0.f, 0.f, 0.f, 0.f, 0.f};
#pragma unroll 8
      for (int k0 = 0; k0 < 384; k0 += 4) {
        int k = k0 + 2 * hl;
        v2f a = Arow[(k0 >> 1) + hl];
        v2f b;
        b.x = W1[(size_t)k * VQ + col0 + lr];
        b.y = W1[(size_t)(k + 1) * VQ + col0 + lr];
        acc = wmma4(a, b, acc);
      }
      float bias = b1[col0 + lr];
#pragma unroll
      for (int i = 0; i < 8; i++) {
        float vv = acc[i] + bias;
        g1[(i + 8 * hl) * HS + col0 + lr] = vv > 0.f ? vv : 0.f;
      }
    }
    __syncthreads();
    // layer 2: g1(16x128) @ W2 -> relu -> g2
    {
      int col0 = wave * 16;
      const v2f* Arow = (const v2f*)(g1 + lr * HS);
      v8f acc = {0.f, 0.f, 0.f, 0.f, 0.f, 0.f, 0.f, 0.f};
#pragma unroll 8
      for (int k0 = 0; k0 < VQ; k0 += 4) {
        int k = k0 + 2 * hl;
        v2f a = Arow[(k0 >> 1) + hl];
        v2f b;
        b.x = W2[(size_t)k * VQ + col0 + lr];
        b.y = W2[(size_t)(k + 1) * VQ + col0 + lr];
        acc = wmma4(a, b, acc);
      }
      float bias = b2[col0 + lr];
#pragma unroll
      for (int i = 0; i < 8; i++) {
        float vv = acc[i] + bias;
        g2[(i + 8 * hl) * HS + col0 + lr] = vv > 0.f ? vv : 0.f;
      }
    }
    __syncthreads();
    // layer 3 + state update (wave 0)
    if (wave == 0) {
      float p = 0.f;
      const float* hrow = g2 + lr * HS + hl * 64;
#pragma unroll 8
      for (int k = 0; k < 64; ++k) p += hrow[k] * W3[hl * 64 + k];
      p += __shfl_xor(p, 16, 32);
      if (hl == 0) {
        float nc = p + b3[0];
        beta_out[lr * SQ + t] = nc;
        Cstate[(size_t)lr * NCQ + cidx[lr]] = nc;
      }
      __threadfence();
    }
    __syncthreads();
  }
}

// ---------------- Kernel F: stream C_seq (327 MB, bandwidth-bound) --------
__global__ __launch_bounds__(256) void k_cseq(const int* __restrict__ c_seq,
                                              const float* __restrict__ beta,
                                              float* __restrict__ Cq) {
  __shared__ int cs[SQ];
  __shared__ float bs[SQ];
  const int nblk = 10;                       // ceil(10000 / 1024)
  int b = blockIdx.x / nblk;
  int cb = blockIdx.x - b * nblk;
  int tid = threadIdx.x;
  for (int i = tid; i < SQ; i += 256) {
    cs[i] = c_seq[b * SQ + i];
    bs[i] = beta[b * SQ + i];
  }
  __syncthreads();
  int c0 = cb * 1024 + tid;
  float v0 = 0.f, v1 = 0.f, v2 = 0.f, v3 = 0.f;
  bool p0 = c0 < NCQ, p1 = c0 + 256 < NCQ, p2 = c0 + 512 < NCQ,
       p3 = c0 + 768 < NCQ;
  for (int t = 0; t < SQ; ++t) {
    int ct = cs[t];
    float bt = bs[t];
    if (ct == c0) v0 = bt;
    if (ct == c0 + 256) v1 = bt;
    if (ct == c0 + 512) v2 = bt;
    if (ct == c0 + 768) v3 = bt;
    size_t base = ((size_t)b * SQ + t) * NCQ + c0;
    if (p0) Cq[base] = v0;
    if (p1) Cq[base + 256] = v1;
    if (p2) Cq[base + 512] = v2;
    if (p3) Cq[base + 768] = v3;
  }
}

// --------------------------------------------------------------------------
extern "C" void kernel_launch(void* const* d_in, const int* in_sizes, int n_in,
                              void* d_out, int out_size, void* d_ws,
                              size_t ws_size, hipStream_t stream) {
  const int* c_seq = (const int*)d_in[0];
  const int* d_seq = (const int*)d_in[1];
  const float* r_seq = (const float*)d_in[2];
  const float* D1 = (const float*)d_in[3];
  const float* D2 = (const float*)d_in[4];
  const float* v_r = (const float*)d_in[5];
  const float* v_beta = (const float*)d_in[6];
  const float* W_ih = (const float*)d_in[7];
  const float* W_hh = (const float*)d_in[8];
  const float* b_ih = (const float*)d_in[9];
  const float* b_hh = (const float*)d_in[10];
  const float* m1W1 = (const float*)d_in[11];
  const float* m1b1 = (const float*)d_in[12];
  const float* m1W2 = (const float*)d_in[13];
  const float* m1b2 = (const float*)d_in[14];
  const float* m1W3 = (const float*)d_in[15];
  const float* m1b3 = (const float*)d_in[16];
  const float* m2W1 = (const float*)d_in[17];
  const float* m2b1 = (const float*)d_in[18];
  const float* m2W2 = (const float*)d_in[19];
  const float* m2b2 = (const float*)d_in[20];
  const float* m2W3 = (const float*)d_in[21];
  const float* m2b3 = (const float*)d_in[22];

  float* out = (float*)d_out;
  float* alpha = out;                              // [B,S]
  float* beta = out + BQ * SQ;                     // [B,S]
  float* gamma = out + 2 * BQ * SQ;                // [B,S]
  float* hseq = out + 3 * BQ * SQ;                 // [B,S,V]
  float* cseq = hseq + (size_t)BQ * SQ * VQ;       // [B,S,NC]

  float* ws = (float*)d_ws;
  float* X = ws;                                   // [8192,256]
  float* XP = X + (size_t)BQ * SQ * 256;           // [8192,384]
  float* Cst = XP + (size_t)BQ * SQ * 384;         // [16,10000]

  k_embed<<<1024, 256, 0, stream>>>(d_seq, r_seq, D1, D2, v_r, X, gamma);
  k_zero<<<(BQ * NCQ + 255) / 256, 256, 0, stream>>>(Cst, BQ * NCQ);
  k_gemm_xp<<<1536, 256, 0, stream>>>(X, W_ih, b_ih, XP);
  k_gru<<<1, 512, 0, stream>>>(XP, W_hh, b_hh, hseq);
  k_alpha<<<128, 128, 0, stream>>>(hseq, m1W1, m1b1, m1W2, m1b2, m1W3, m1b3,
                                   alpha);
  k_beta<<<1, 256, 0, stream>>>(c_seq, X, v_beta, m2W1, m2b1, m2W2, m2b2, m2W3,
                                m2b3, Cst, beta);
  k_cseq<<<BQ * 10, 256, 0, stream>>>(c_seq, beta, cseq);
}